// HCRLayer_9225589752137
// MI455X (gfx1250) — compile-verified
//
#include <hip/hip_runtime.h>

typedef float v2f __attribute__((ext_vector_type(2)));
typedef float v8f __attribute__((ext_vector_type(8)));

#define ROWS_PER_WAVE 32
#define ROWS_PER_BLOCK 128   // 4 waves * 32 rows
#define SLOT 74              // floats per row slot: u[0..35], pad[36..37], v[38..73]

__global__ __launch_bounds__(128)
void hcr_wmma_kernel(const float* __restrict__ x,      // (B, 4)
                     const float* __restrict__ C,      // (32, 1296) row-major
                     float* __restrict__ out)          // (B, 32)
{
    __shared__ float smem[ROWS_PER_BLOCK * SLOT];      // 37.9 KB

    const int tid  = threadIdx.x;
    const int lane = tid & 31;
    const int wid  = tid >> 5;
    const int hs   = lane >> 4;        // half-wave select
    const int ln   = lane & 15;

    const size_t blockBase = (size_t)blockIdx.x * ROWS_PER_BLOCK;

    // ---------------- Phase 1: Legendre basis + pairwise products into LDS ----------
    // lanes 0-15 build u = b0 (x) b1 ; lanes 16-31 build v = b2 (x) b3.
    // Each half-wave covers 16 rows per pass; two passes cover the wave's 32 rows.
    const float scale0 = 1.0f;
    const float scale1 = 1.7320508075688772f;  // sqrt(3)
    const float scale2 = 2.2360679774997896f;  // sqrt(5)
    const float scale3 = 2.6457513110645907f;  // sqrt(7)
    const float scale4 = 3.0f;                 // sqrt(9)
    const float scale5 = 3.3166247903553998f;  // sqrt(11)

    #pragma unroll
    for (int t = 0; t < 2; ++t) {
        const int rid = wid * ROWS_PER_WAVE + ln + 16 * t;     // row within block
        const float* xp = x + (blockBase + rid) * 4 + 2 * hs;
        const float xa = xp[0];
        const float xb = xp[1];
        const float ta = 2.0f * xa - 1.0f;
        const float tb = 2.0f * xb - 1.0f;

        float Pa0 = 1.0f, Pa1 = ta;
        float Pb0 = 1.0f, Pb1 = tb;
        float Pa2 = (3.0f * ta * Pa1 - 1.0f * Pa0) * 0.5f;
        float Pb2 = (3.0f * tb * Pb1 - 1.0f * Pb0) * 0.5f;
        float Pa3 = (5.0f * ta * Pa2 - 2.0f * Pa1) * (1.0f / 3.0f);
        float Pb3 = (5.0f * tb * Pb2 - 2.0f * Pb1) * (1.0f / 3.0f);
        float Pa4 = (7.0f * ta * Pa3 - 3.0f * Pa2) * 0.25f;
        float Pb4 = (7.0f * tb * Pb3 - 3.0f * Pb2) * 0.25f;
        float Pa5 = (9.0f * ta * Pa4 - 4.0f * Pa3) * 0.2f;
        float Pb5 = (9.0f * tb * Pb4 - 4.0f * Pb3) * 0.2f;

        float ba[6] = { Pa0 * scale0, Pa1 * scale1, Pa2 * scale2,
                        Pa3 * scale3, Pa4 * scale4, Pa5 * scale5 };
        float bb[6] = { Pb0 * scale0, Pb1 * scale1, Pb2 * scale2,
                        Pb3 * scale3, Pb4 * scale4, Pb5 * scale5 };

        float* slot = smem + rid * SLOT + hs * 38;   // u at +0, v at +38
        #pragma unroll
        for (int i = 0; i < 6; ++i) {
            #pragma unroll
            for (int j = 0; j < 6; ++j)
                slot[i * 6 + j] = ba[i] * bb[j];
        }
    }
    __syncthreads();

    // ---------------- Phase 2: K=1296 contraction via V_WMMA_F32_16X16X4_F32 --------
    // A fragment (16x4 f32, 2 VGPRs): lane ln holds row M=ln; VGPR0 = K row (2*hs),
    // VGPR1 = K row (2*hs + 1).  A[M, kk] = u[M, kk/36] * v[M, kk%36].
    // B fragment (4x16 f32, 2 VGPRs): lane holds column N=ln; K rows 2*hs, 2*hs+1
    // in VGPR0/1 -> contiguous b64 load from C[N, kk..kk+1].
    const int krow = 2 * hs;
    const float* uRow0 = smem + (wid * ROWS_PER_WAVE + ln) * SLOT;   // M-tile 0 row
    const float* uRow1 = uRow0 + 16 * SLOT;                          // M-tile 1 row
    const float* vRow0 = uRow0 + 38;
    const float* vRow1 = uRow1 + 38;
    const float* Bq0 = C + (size_t)ln * 1296 + krow;                 // N-tile 0
    const float* Bq1 = C + (size_t)(ln + 16) * 1296 + krow;          // N-tile 1

    v8f acc00 = {}, acc01 = {}, acc10 = {}, acc11 = {};

    for (int p = 0; p < 36; ++p) {               // p = kk/36, wave-uniform (36 % 4 == 0)
        const float up0 = uRow0[p];
        const float up1 = uRow1[p];
        const int   kb  = p * 36;
        #pragma unroll 3
        for (int j = 0; j < 36; j += 4) {        // q = j + krow never wraps within chunk
            const int q = j + krow;
            const v2f vv0 = *(const v2f*)(vRow0 + q);      // ds_load_b64, 8B aligned
            const v2f vv1 = *(const v2f*)(vRow1 + q);
            const v2f b0  = *(const v2f*)(Bq0 + kb + j);   // global_load_b64 (L2-hot)
            const v2f b1  = *(const v2f*)(Bq1 + kb + j);
            v2f a0; a0.x = up0 * vv0.x; a0.y = up0 * vv0.y;
            v2f a1; a1.x = up1 * vv1.x; a1.y = up1 * vv1.y;
            acc00 = __builtin_amdgcn_wmma_f32_16x16x4_f32(false, a0, false, b0,
                                                          (short)0, acc00, false, false);
            acc01 = __builtin_amdgcn_wmma_f32_16x16x4_f32(false, a0, false, b1,
                                                          (short)0, acc01, false, false);
            acc10 = __builtin_amdgcn_wmma_f32_16x16x4_f32(false, a1, false, b0,
                                                          (short)0, acc10, false, false);
            acc11 = __builtin_amdgcn_wmma_f32_16x16x4_f32(false, a1, false, b1,
                                                          (short)0, acc11, false, false);
        }
    }

    // ---------------- Phase 3: epilogue  out = 1 + full - c_flat[:,0] ---------------
    // C/D layout: VGPR r -> lanes 0-15: (M=r, N=lane); lanes 16-31: (M=8+r, N=lane-16).
    const float z0 = C[(size_t)ln * 1296];          // c_flat[n0, 0]
    const float z1 = C[(size_t)(ln + 16) * 1296];   // c_flat[n0+16, 0]
    float* op = out + (blockBase + wid * ROWS_PER_WAVE + 8 * hs) * 32 + ln;
    #pragma unroll
    for (int r = 0; r < 8; ++r) {
        op[r * 32]              = 1.0f + acc00[r] - z0;
        op[r * 32 + 16]         = 1.0f + acc01[r] - z1;
        op[(r + 16) * 32]       = 1.0f + acc10[r] - z0;
        op[(r + 16) * 32 + 16]  = 1.0f + acc11[r] - z1;
    }
}

extern "C" void kernel_launch(void* const* d_in, const int* in_sizes, int n_in,
                              void* d_out, int out_size, void* d_ws, size_t ws_size,
                              hipStream_t stream) {
    const float* x = (const float*)d_in[0];   // (B, 4) float32
    const float* C = (const float*)d_in[1];   // (32, 6,6,6,6) -> (32, 1296) float32
    float* out = (float*)d_out;               // (B, 32) float32

    const int B = in_sizes[0] / 4;            // 131072
    const int blocks = B / ROWS_PER_BLOCK;    // 1024
    hcr_wmma_kernel<<<blocks, 128, 0, stream>>>(x, C, out);
}